// PCENLayer_11793980195071
// MI455X (gfx1250) — compile-verified
//
#include <hip/hip_runtime.h>
#include <math.h>
#include <stdint.h>

// ---------------------------------------------------------------------------
// PCEN (filtfilt first-order IIR + pointwise) for MI455X / gfx1250.
//
// Forward pass:  y = L @ x per 16-chunk tile, L lower-triangular Toeplitz,
//                L[i][k] = b0*c^(i-k);  chunk state H[ch+1]=c^16 H[ch]+S15[ch]
//                -> forward weighted Hillis-Steele scan.
// Backward pass: z = U @ y, U upper-triangular Toeplitz U[i][k]=b0*c^(k-i);
//                chunk state G[ch]=c^16 G[ch+1]+S0[ch] -> REVERSE weighted
//                scan. No reversed indexing anywhere: both passes use the
//                same forward-ordered, guard-free, vectorized LDS accesses.
// Each 16x16x16 chunk-tile matmul = 4x V_WMMA_F32_16X16X4_F32 (f32 kept for
// IIR accuracy over 16000 steps). Staging global->LDS uses
// GLOBAL_LOAD_ASYNC_TO_LDS_B128 (ASYNCcnt) - 512B/wave/instruction.
//
// Traffic: 131MB x + 131MB x re-read + 131MB out ~= 393MB @ 23.3TB/s ~ 17us.
// 2048 blocks x 8 waves; ~79KB LDS/block -> 4 blocks/WGP (320KB).
// ---------------------------------------------------------------------------

typedef __attribute__((ext_vector_type(2))) float v2f;
typedef __attribute__((ext_vector_type(8))) float v8f;

#define T_LEN   16000
#define NTILE   63                 // ceil(16000/256)
#define TP      (NTILE * 256)      // 16128 padded samples
#define NCHR    1024               // scan length (pow2 >= 1008 chunks)
#define GUARD   512                // zero guard on both sides of scan bufs
#define CBUF    (GUARD + NCHR + GUARD)
#define LOG_EPS_F (-13.815510557964274f)   // log(1e-6)

struct __align__(16) SmemT {
    float xs[TP];        // signal row: x -> S -> y -> S' -> M   (64512 B)
    float ca[CBUF];      // carry / scan ping (guarded)          ( 8192 B)
    float cb[CBUF];      // scan pong (guarded)                  ( 8192 B)
    float cpow[17];      // c^0 .. c^16
    float dummy;         // sink for inactive-half carry stores
};

// One lfilter pass over sm->xs, in place. BWD runs the time-reversed
// recurrence *without* reversed indexing (upper-triangular A, reverse scan).
template <bool BWD>
__device__ __forceinline__ void iir_pass(SmemT* sm, float b0, float cc,
                                         float inva0, int tid)
{
    const int  lane   = tid & 31;
    const int  wave   = tid >> 5;
    const int  col    = lane & 15;     // chunk column held by this lane
    const bool hiHalf = lane >= 16;

    // Boundary samples (kept raw by the reference's init).
    float w0 = sm->xs[BWD ? (T_LEN - 1) : 0];
    float w1 = sm->xs[BWD ? (T_LEN - 2) : 1];

    // Zero real carry region; before BWD also zero the 128-sample pad
    // (pad chunks precede real chunks in the reverse scan direction).
    for (int i = tid; i < NCHR; i += 256) sm->ca[GUARD + i] = 0.0f;
    if (BWD && tid < (TP - T_LEN)) sm->xs[T_LEN + tid] = 0.0f;
    __syncthreads();

    // A fragments for V_WMMA_F32_16X16X4_F32 (lanes 0-15: K=kb*4+{0,1},
    // lanes 16-31: K=kb*4+{2,3}).  FWD: L[i][k]=b0 c^(i-k) (i>=k);
    // BWD: U[i][k]=b0 c^(k-i) (k>=i).
    v2f afrag[4];
    for (int kb = 0; kb < 4; ++kb) {
        int k0 = kb * 4 + (hiHalf ? 2 : 0);
        int d0 = BWD ? (k0 - col)     : (col - k0);
        int d1 = BWD ? (k0 + 1 - col) : (col - k0 - 1);
        afrag[kb][0] = (d0 >= 0) ? b0 * sm->cpow[d0] : 0.0f;
        afrag[kb][1] = (d1 >= 0) ? b0 * sm->cpow[d1] : 0.0f;
    }

    // ---- Tile phase: S = A @ X, guard-free & vectorized ----
    for (int tile = wave; tile < NTILE; tile += 8) {   // wave-uniform
        const int base = tile * 256 + col * 16;

        v2f bfrag[4];                                  // ds_load_b64 x4
        for (int kb = 0; kb < 4; ++kb) {
            int r0 = kb * 4 + (hiHalf ? 2 : 0);
            bfrag[kb] = *(const v2f*)&sm->xs[base + r0];
        }

        v8f acc = {};
        for (int kb = 0; kb < 4; ++kb)
            acc = __builtin_amdgcn_wmma_f32_16x16x4_f32(
                false, afrag[kb], false, bfrag[kb], (short)0, acc,
                false, false);

        // Write S rows back: 8 contiguous floats per lane, 16B aligned.
        const int rb0 = hiHalf ? 8 : 0;
        *(float4*)&sm->xs[base + rb0]     = make_float4(acc[0], acc[1],
                                                        acc[2], acc[3]);
        *(float4*)&sm->xs[base + rb0 + 4] = make_float4(acc[4], acc[5],
                                                        acc[6], acc[7]);

        // Chunk carry: FWD = row15 (hi-half acc[7]); BWD = row0 (lo acc[0]).
        // Inactive half stores to a dummy word: select address, no branch.
        float  cval = BWD ? acc[0] : acc[7];
        bool   act  = BWD ? (!hiHalf) : hiHalf;
        float* cp   = act ? &sm->ca[GUARD + tile * 16 + col] : &sm->dummy;
        *cp = cval;
    }
    __syncthreads();

    // Initial-state correction folds into the scan seed:
    //   K = (1-b0)*w1 - c*b0*w0,  Hb = K / c^2.
    float Kc = (1.0f - b0) * w1 - cc * b0 * w0;
    float Hb = Kc / (cc * cc);
    float q  = sm->cpow[16];                        // c^16
    if (tid == 0) {
        int seed = BWD ? 999 : 0;                   // 1000 real chunks
        sm->ca[GUARD + seed] = fmaf(q, Hb, sm->ca[GUARD + seed]);
    }

    // ---- Weighted Hillis-Steele scan (ratio q), branch-free via guards ----
    float  mult = q;
    float* src  = sm->ca + GUARD;
    float* dst  = sm->cb + GUARD;
    for (int d = 1; d < NCHR; d <<= 1) {            // 10 rounds -> ends in ca
        __syncthreads();
        for (int i = tid; i < NCHR; i += 256) {
            float nb = BWD ? src[i + d] : src[i - d];   // guard reads 0
            dst[i] = fmaf(mult, nb, src[i]);
        }
        float* t = src; src = dst; dst = t;
        mult *= mult;
    }
    __syncthreads();                                 // result in src (== ca)

    // ---- Apply: FWD y[t]=S[i]+c^(i+1)H[ch]; BWD z[t]=S'[i]+c^(16-i)G[ch+1]
    for (int t = tid; t < TP; t += 256) {            // full padded range
        int   ch = t >> 4, i = t & 15;
        float g  = BWD ? src[ch + 1] : src[ch - 1];  // guard reads 0 @ ch==0
        float Hn = BWD ? ((ch == 999) ? Hb : g)
                       : ((ch == 0)   ? Hb : g);
        float cp = sm->cpow[BWD ? (16 - i) : (i + 1)];
        sm->xs[t] = fmaf(cp, Hn, sm->xs[t]) * inva0;
    }
    __syncthreads();
    if (tid == 0) {                                  // forced endpoints
        if (BWD) { sm->xs[T_LEN - 1] = w0 * inva0; sm->xs[T_LEN - 2] = w1 * inva0; }
        else     { sm->xs[0]         = w0 * inva0; sm->xs[1]         = w1 * inva0; }
    }
    __syncthreads();
}

__global__ void __launch_bounds__(256)
pcen_kernel(const float* __restrict__ x,
            const float* __restrict__ i_sig_alpha,
            const float* __restrict__ log_delta,
            const float* __restrict__ i_sig_r,
            const float* __restrict__ b,
            const float* __restrict__ a,
            float* __restrict__ out,
            int F)
{
    __shared__ SmemT sm;
    const int       tid = threadIdx.x;
    const int       row = blockIdx.x;
    const long long rb  = (long long)row * T_LEN;
    const int       f   = row % F;

    const float b0    = b[0];
    const float cc    = -a[1];                 // y[t] = b0 x[t] + c y[t-1]
    const float inva0 = 1.0f / a[0];
    const float al    = 1.0f / (1.0f + expf(-i_sig_alpha[f]));
    const float delta = expf(log_delta[f]);
    const float rr    = 1.0f / (1.0f + expf(-i_sig_r[f]));
    const float dr    = powf(delta, rr);

    // Zero scan guard zones once (never written afterwards) + c-power table.
    for (int i = tid; i < GUARD; i += 256) {
        sm.ca[i] = 0.0f; sm.ca[GUARD + NCHR + i] = 0.0f;
        sm.cb[i] = 0.0f; sm.cb[GUARD + NCHR + i] = 0.0f;
    }
    if (tid < 17) sm.cpow[tid] = powf(cc, (float)tid);

    // ---- Stage the row into LDS with async b128 copies (ASYNCcnt) ----
    {
        // Low 32 bits of a flat shared pointer are the LDS byte offset.
        unsigned           xs_lds = (unsigned)(uintptr_t)&sm.xs[0];
        unsigned long long gaddr  = (unsigned long long)(uintptr_t)(x + rb);
        for (int i4 = tid; i4 < T_LEN / 4; i4 += 256) {
            unsigned           lo = xs_lds + (unsigned)(i4 * 16);
            unsigned long long ga = gaddr + (unsigned long long)i4 * 16;
            asm volatile("global_load_async_to_lds_b128 %0, %1, off"
                         :: "v"(lo), "v"(ga) : "memory");
        }
        asm volatile("s_wait_asynccnt 0x0" ::: "memory");
    }
    __syncthreads();

    iir_pass<false>(&sm, b0, cc, inva0, tid);  // forward lfilter
    iir_pass<true >(&sm, b0, cc, inva0, tid);  // backward lfilter (filtfilt)

    // ---- Pointwise PCEN, float4-vectorized ----
    const float4* x4 = (const float4*)(x + rb);
    const float4* m4 = (const float4*)sm.xs;
    float4*       o4 = (float4*)(out + rb);
    for (int i4 = tid; i4 < T_LEN / 4; i4 += 256) {
        float4 xv = x4[i4];
        float4 mv = m4[i4];
        float4 r4;
        {
            float M;
            M = __expf(-al * (LOG_EPS_F + __logf(fmaf(mv.x, 1e6f, 1.0f))));
            r4.x = __expf(rr * __logf(fmaf(xv.x, M, delta))) - dr;
            M = __expf(-al * (LOG_EPS_F + __logf(fmaf(mv.y, 1e6f, 1.0f))));
            r4.y = __expf(rr * __logf(fmaf(xv.y, M, delta))) - dr;
            M = __expf(-al * (LOG_EPS_F + __logf(fmaf(mv.z, 1e6f, 1.0f))));
            r4.z = __expf(rr * __logf(fmaf(xv.z, M, delta))) - dr;
            M = __expf(-al * (LOG_EPS_F + __logf(fmaf(mv.w, 1e6f, 1.0f))));
            r4.w = __expf(rr * __logf(fmaf(xv.w, M, delta))) - dr;
        }
        o4[i4] = r4;
    }
}

extern "C" void kernel_launch(void* const* d_in, const int* in_sizes, int n_in,
                              void* d_out, int out_size, void* d_ws, size_t ws_size,
                              hipStream_t stream)
{
    const float* x   = (const float*)d_in[0];
    const float* isa = (const float*)d_in[1];
    const float* ld  = (const float*)d_in[2];
    const float* isr = (const float*)d_in[3];
    const float* b   = (const float*)d_in[4];
    const float* a   = (const float*)d_in[5];
    float* out = (float*)d_out;

    const int F     = in_sizes[1];             // 128 bands
    const int nrows = in_sizes[0] / T_LEN;     // B*C*F = 2048 rows

    pcen_kernel<<<dim3(nrows), dim3(256), 0, stream>>>(x, isa, ld, isr, b, a,
                                                       out, F);
}